// ProbAttention_57294863729124
// MI455X (gfx1250) — compile-verified
//
#include <hip/hip_runtime.h>
#include <cmath>

// Problem constants (match the harness setup exactly)
constexpr int Bc = 4, Lc = 2048, Hc = 8, Dc = 64;
constexpr int HD = Hc * Dc;          // 512
constexpr int NCH = 32, CHUNK = Lc / NCH;  // cumsum chunks
constexpr int NW = 4, KPW = Lc / NW; // waves per flash block, keys per wave

typedef float v2f __attribute__((ext_vector_type(2)));
typedef float v8f __attribute__((ext_vector_type(8)));

static __device__ __forceinline__ v8f wmma4(v2f a, v2f b, v8f c) {
#if defined(__HIP_DEVICE_COMPILE__)
#if __has_builtin(__builtin_amdgcn_wmma_f32_16x16x4_f32)
  // (neg_a, A, neg_b, B, c_mod, C, reuse_a, reuse_b)
  return __builtin_amdgcn_wmma_f32_16x16x4_f32(false, a, false, b, (short)0, c,
                                               false, false);
#else
#error "no fp32 WMMA builtin on gfx1250 device pass"
#endif
#else
  // Host pass: never executed; stub so the TU parses.
  (void)a; (void)b;
  return c;
#endif
}

// ---------------------------------------------------------------------------
// Kernel A: sampled sparsity measure M[b,h,l] = max_s(QK) - sum_s(QK)/L
// One wave per (b,h,l). Lanes split D=64 as 2 floats each.
// ---------------------------------------------------------------------------
__global__ __launch_bounds__(256) void pa_scores(
    const float* __restrict__ Q, const float* __restrict__ K,
    const int* __restrict__ idx, float* __restrict__ Mout, int S) {
  int wid = (blockIdx.x * blockDim.x + threadIdx.x) >> 5;
  int lane = threadIdx.x & 31;
  if (wid >= Bc * Hc * Lc) return;
  int l = wid % Lc;
  int bh = wid / Lc;
  int h = bh % Hc, b = bh / Hc;
  const float* qp = Q + ((size_t)b * Lc + l) * HD + h * Dc;
  float q0 = qp[lane], q1 = qp[lane + 32];
  float mx = -INFINITY, sm = 0.f;
  const int* ip = idx + (size_t)l * S;
  for (int s = 0; s < S; ++s) {
    int kr = ip[s];
    const float* kp = K + ((size_t)b * Lc + kr) * HD + h * Dc;
    float p = q0 * kp[lane] + q1 * kp[lane + 32];
    p += __shfl_xor(p, 16, 32);
    p += __shfl_xor(p, 8, 32);
    p += __shfl_xor(p, 4, 32);
    p += __shfl_xor(p, 2, 32);
    p += __shfl_xor(p, 1, 32);
    mx = fmaxf(mx, p);
    sm += p;
  }
  if (lane == 0) Mout[(size_t)bh * Lc + l] = mx - sm / (float)Lc;
}

// ---------------------------------------------------------------------------
// Kernel B: top-k (k=nt) per (b,h) by iterative argmax, ties -> lowest index
// ---------------------------------------------------------------------------
__global__ __launch_bounds__(256) void pa_topk(const float* __restrict__ M,
                                               int* __restrict__ top, int nt) {
  int bh = blockIdx.x, tid = threadIdx.x;
  __shared__ float sv[Lc];
  __shared__ float rv[8];
  __shared__ int ri[8];
  for (int i = tid; i < Lc; i += blockDim.x) sv[i] = M[(size_t)bh * Lc + i];
  __syncthreads();
  for (int t = 0; t < nt; ++t) {
    float bv = -INFINITY;
    int bi = 0x7fffffff;
    for (int i = tid; i < Lc; i += blockDim.x) {
      float v = sv[i];
      if (v > bv || (v == bv && i < bi)) { bv = v; bi = i; }
    }
#pragma unroll
    for (int m = 16; m >= 1; m >>= 1) {
      float ov = __shfl_xor(bv, m, 32);
      int oi = __shfl_xor(bi, m, 32);
      if (ov > bv || (ov == bv && oi < bi)) { bv = ov; bi = oi; }
    }
    if ((tid & 31) == 0) { rv[tid >> 5] = bv; ri[tid >> 5] = bi; }
    __syncthreads();
    if (tid == 0) {
      float gv = -INFINITY;
      int gi = 0x7fffffff;
      for (int wv = 0; wv < (int)(blockDim.x >> 5); ++wv)
        if (rv[wv] > gv || (rv[wv] == gv && ri[wv] < gi)) { gv = rv[wv]; gi = ri[wv]; }
      top[(size_t)bh * nt + t] = gi;
      sv[gi] = -INFINITY;
    }
    __syncthreads();
  }
}

// ---------------------------------------------------------------------------
// Kernels C1/C2: cumsum(V, axis=L) written transposed into out[b,l,h,d]
// ---------------------------------------------------------------------------
__global__ __launch_bounds__(Dc) void pa_cumsum1(const float* __restrict__ V,
                                                 float* __restrict__ out,
                                                 float* __restrict__ csum) {
  int chunk = blockIdx.x % NCH, bh = blockIdx.x / NCH;
  int h = bh % Hc, b = bh / Hc, d = threadIdx.x;
  size_t base = ((size_t)b * Lc + chunk * CHUNK) * HD + h * Dc + d;
  float run = 0.f;
  for (int i = 0; i < CHUNK; ++i) {
    run += V[base + (size_t)i * HD];
    out[base + (size_t)i * HD] = run;
  }
  csum[((size_t)bh * NCH + chunk) * Dc + d] = run;
}

__global__ __launch_bounds__(Dc) void pa_cumsum2(float* __restrict__ out,
                                                 const float* __restrict__ csum) {
  int chunk = blockIdx.x % NCH, bh = blockIdx.x / NCH;
  int h = bh % Hc, b = bh / Hc, d = threadIdx.x;
  if (chunk == 0) return;
  float off = 0.f;
  for (int c = 0; c < chunk; ++c) off += csum[((size_t)bh * NCH + c) * Dc + d];
  size_t base = ((size_t)b * Lc + chunk * CHUNK) * HD + h * Dc + d;
  for (int i = 0; i < CHUNK; ++i) out[base + (size_t)i * HD] += off;
}

// ---------------------------------------------------------------------------
// Kernel D: flash attention over selected queries using V_WMMA_F32_16X16X4_F32
// Block = 4 waves; block handles (b,h, 16-row query tile); each wave covers
// 512 keys with online softmax; partials combined in LDS.
// ---------------------------------------------------------------------------
__global__ __launch_bounds__(128) void pa_flash(
    const float* __restrict__ Q, const float* __restrict__ K,
    const float* __restrict__ V, const int* __restrict__ top,
    float* __restrict__ out, int nt, int MT) {
  int mt = blockIdx.x % MT, bh = blockIdx.x / MT;
  int h = bh % Hc, b = bh / Hc;
  int tid = threadIdx.x, w = tid >> 5, lane = tid & 31;
  int l16 = lane & 15, half = lane >> 4, koff = half * 2;

  __shared__ int s_rows[16];
  __shared__ float s_P[NW][256];
  __shared__ float s_O[NW][16][Dc];
  __shared__ float s_m[NW][16];
  __shared__ float s_l[NW][16];

  if (tid < 16) {
    int qi = mt * 16 + tid;
    s_rows[tid] = (qi < nt) ? top[(size_t)bh * nt + qi] : -1;
  }
  __syncthreads();

  // A-fragments of the 16x64 Q tile (16 chunks of K=4)
  int qrow = s_rows[l16];
  float qmask = (qrow >= 0) ? 1.f : 0.f;
  int qr0 = (qrow >= 0) ? qrow : 0;
  const float* qp = Q + ((size_t)b * Lc + qr0) * HD + h * Dc + koff;
  v2f qa[16];
#pragma unroll
  for (int kc = 0; kc < 16; ++kc) {
    v2f t = *(const v2f*)(qp + kc * 4);
    qa[kc].x = t.x * qmask;
    qa[kc].y = t.y * qmask;
  }

  float ms[8], ls[8];
  v8f Oa[4];
#pragma unroll
  for (int r = 0; r < 8; ++r) { ms[r] = -INFINITY; ls[r] = 0.f; }
#pragma unroll
  for (int n = 0; n < 4; ++n) Oa[n] = (v8f){0.f, 0.f, 0.f, 0.f, 0.f, 0.f, 0.f, 0.f};

  const float scale = 0.125f;  // 1/sqrt(64)

  for (int kb = 0; kb < KPW / 16; ++kb) {
    int key0 = w * KPW + kb * 16;

    // GEMM1: S(16x16) = Qtile(16x64) * K^T(64x16), 16 fp32 WMMAs
    v8f c = (v8f){0.f, 0.f, 0.f, 0.f, 0.f, 0.f, 0.f, 0.f};
    const float* kp = K + ((size_t)b * Lc + key0 + l16) * HD + h * Dc + koff;
#pragma unroll
    for (int kc = 0; kc < 16; ++kc) {
      v2f bf = *(const v2f*)(kp + kc * 4);
      c = wmma4(qa[kc], bf, c);
    }

    // Online softmax in the C-layout (row m = r + 8*half, col n = l16)
#pragma unroll
    for (int r = 0; r < 8; ++r) {
      float s = c[r] * scale;
      float t = s;
      t = fmaxf(t, __shfl_xor(t, 1, 32));
      t = fmaxf(t, __shfl_xor(t, 2, 32));
      t = fmaxf(t, __shfl_xor(t, 4, 32));
      t = fmaxf(t, __shfl_xor(t, 8, 32));
      float mnew = fmaxf(ms[r], t);
      float corr = __expf(ms[r] - mnew);
      float p = __expf(s - mnew);
      float rs = p;
      rs += __shfl_xor(rs, 1, 32);
      rs += __shfl_xor(rs, 2, 32);
      rs += __shfl_xor(rs, 4, 32);
      rs += __shfl_xor(rs, 8, 32);
      ls[r] = ls[r] * corr + rs;
      ms[r] = mnew;
      Oa[0][r] *= corr;
      Oa[1][r] *= corr;
      Oa[2][r] *= corr;
      Oa[3][r] *= corr;
      s_P[w][(r + 8 * half) * 16 + l16] = p;  // C-layout -> LDS
    }
    // Re-read P in A-layout (same-wave DS ops are in order; compiler waits)
    v2f pa[4];
#pragma unroll
    for (int kc2 = 0; kc2 < 4; ++kc2)
      pa[kc2] = *(const v2f*)(&s_P[w][l16 * 16 + kc2 * 4 + koff]);

    // GEMM2: O(16x64) += P(16x16) * Vblock(16x64), 16 fp32 WMMAs
    const float* vbase = V + ((size_t)b * Lc + key0) * HD + h * Dc;
#pragma unroll
    for (int n = 0; n < 4; ++n) {
#pragma unroll
      for (int kc2 = 0; kc2 < 4; ++kc2) {
        const float* vp = vbase + (size_t)(kc2 * 4 + koff) * HD + n * 16 + l16;
        v2f bf;
        bf.x = vp[0];
        bf.y = vp[HD];
        Oa[n] = wmma4(pa[kc2], bf, Oa[n]);
      }
    }
  }

  // Publish per-wave partials
#pragma unroll
  for (int r = 0; r < 8; ++r) {
    int row = r + 8 * half;
    if (l16 == 0) { s_m[w][row] = ms[r]; s_l[w][row] = ls[r]; }
#pragma unroll
    for (int n = 0; n < 4; ++n) s_O[w][row][n * 16 + l16] = Oa[n][r];
  }
  __syncthreads();

  // Combine 4 partial softmaxes, normalize, scatter to selected rows
  for (int e = tid; e < 16 * Dc; e += blockDim.x) {
    int row = e >> 6, d = e & (Dc - 1);
    float Mx = s_m[0][row];
    for (int w2 = 1; w2 < NW; ++w2) Mx = fmaxf(Mx, s_m[w2][row]);
    float Ls = 0.f, o = 0.f;
    for (int w2 = 0; w2 < NW; ++w2) {
      float f = __expf(s_m[w2][row] - Mx);
      Ls += s_l[w2][row] * f;
      o += s_O[w2][row][d] * f;
    }
    int qi = mt * 16 + row;
    if (qi < nt) {
      int rq = s_rows[row];
      out[((size_t)b * Lc + rq) * HD + h * Dc + d] = o / Ls;
    }
  }
}

// ---------------------------------------------------------------------------
extern "C" void kernel_launch(void* const* d_in, const int* in_sizes, int n_in,
                              void* d_out, int out_size, void* d_ws,
                              size_t ws_size, hipStream_t stream) {
  const float* Q = (const float*)d_in[0];
  const float* K = (const float*)d_in[1];
  const float* V = (const float*)d_in[2];
  const int* idx = (const int*)d_in[3];
  float* out = (float*)d_out;

  int S = in_sizes[3] / Lc;  // sample_k (== n_top by construction)
  int nt = S;

  float* wsf = (float*)d_ws;
  float* Mbuf = wsf;                              // B*H*L floats
  float* csum = wsf + (size_t)Bc * Hc * Lc;       // B*H*NCH*D floats
  int* top = (int*)(csum + (size_t)Bc * Hc * NCH * Dc);  // B*H*nt ints

  dim3 gA((unsigned)((Bc * Hc * Lc * 32) / 256));
  pa_scores<<<gA, 256, 0, stream>>>(Q, K, idx, Mbuf, S);
  pa_topk<<<Bc * Hc, 256, 0, stream>>>(Mbuf, top, nt);
  pa_cumsum1<<<Bc * Hc * NCH, Dc, 0, stream>>>(V, out, csum);
  pa_cumsum2<<<Bc * Hc * NCH, Dc, 0, stream>>>(out, csum);
  int MT = (nt + 15) / 16;
  pa_flash<<<Bc * Hc * MT, 128, 0, stream>>>(Q, K, V, top, out, nt, MT);
}